// Topm_CrossAttention_Restormer_41996190220657
// MI455X (gfx1250) — compile-verified
//
#include <hip/hip_runtime.h>
#include <hip/hip_bf16.h>
#include <cstdint>

// ---------------------------------------------------------------------------
// Problem constants (fixed shapes from the reference)
// ---------------------------------------------------------------------------
#define BATCH   8
#define CDIM    384
#define HEADS   8
#define CH      48          // channels per head
#define IMGH    128
#define IMGW    128
#define NPIX    16384       // 128*128

typedef __attribute__((ext_vector_type(2))) float v2f;
typedef __attribute__((ext_vector_type(8))) float v8f;
typedef __attribute__((ext_vector_type(4))) uint32_t u32x4;
typedef __attribute__((ext_vector_type(8))) uint32_t u32x8;

// ---------------------------------------------------------------------------
// Tensor Data Mover: post a 2D tile load (global -> LDS) via D# descriptor.
// Group0: count=1 | lds_addr | global_addr(57b) | type=2 ("image").
// Group1: data_size=4B, LDS padding of 4 DWORDs after every 64 DWORDs
//         (reproduces the 68-float bank-conflict-free row pitch),
//         tensor dims (for OOB), tile dims, dim0 stride.
// 2-group form: VADDR2/VADDR3 = NULL -> tensors up to 2D.
// Issued by one wave; completion tracked with TENSORcnt.
// ---------------------------------------------------------------------------
__device__ __forceinline__ void tdm_load_tile_2d(
    uint32_t lds_addr, const float* gptr,
    uint32_t tensor_d0, uint32_t tensor_d1,
    uint32_t tile_d0, uint32_t tile_d1, uint32_t stride0_elems)
{
  uint64_t ga = (uint64_t)(uintptr_t)gptr;
  u32x4 g0;
  g0[0] = 1u;                                        // count=1, user mode
  g0[1] = lds_addr;                                  // LDS byte address
  g0[2] = (uint32_t)ga;                              // global_addr[31:0]
  g0[3] = (uint32_t)(ga >> 32) | (2u << 30);         // global_addr[56:32] | type=2
  u32x8 g1;
  g1[0] = (2u << 16)        // data_size = 4 bytes
        | (1u << 20)        // pad_enable
        | (5u << 22)        // pad_interval: 64 DWORDs
        | (3u << 25);       // pad_amount: 4 DWORDs
  g1[1] = (tensor_d0 & 0xFFFFu) << 16;                       // tensor_dim0[15:0]
  g1[2] = (tensor_d0 >> 16) | ((tensor_d1 & 0xFFFFu) << 16); // d0[31:16] | d1[15:0]
  g1[3] = (tensor_d1 >> 16) | (tile_d0 << 16);               // d1[31:16] | tile_dim0
  g1[4] = tile_d1;                                           // tile_dim1 | tile_dim2=0
  g1[5] = stride0_elems;                                     // tensor_dim0_stride[31:0]
  g1[6] = 0;
  g1[7] = 0;
  asm volatile("tensor_load_to_lds %0, %1" :: "s"(g0), "s"(g1) : "memory");
}

// ---------------------------------------------------------------------------
// Kernel 1: 1x1 conv as GEMM  Y[b,o,n] = sum_c W[o,c] * X[b,c,n]
// WMMA f32 16x16x4. Block = 256 threads = 8 waves arranged 4(M) x 2(N),
// each wave owns a 16x32 output patch -> block tile = 64(M) x 64(N).
// B tiles (activations, the bandwidth consumer) are streamed by the TDM into
// double-buffered LDS, overlapped with WMMA compute via TENSORcnt.
// ---------------------------------------------------------------------------
#define GT_M 64
#define GT_N 64
#define GT_K 32
#define A_PAD 36   // 144B rows: 16B aligned, conflict-free column reads
#define B_PAD 68   // 272B rows: matches TDM pad (64+4 DWORDs), conflict-free

__global__ __launch_bounds__(256) void gemm1x1_wmma_kernel(
    const float* __restrict__ X,   // [B, Cin, NPIX]
    const float* __restrict__ W,   // [O, Cin]
    float* __restrict__ Y,         // [B, O, NPIX]
    int Cin, int O)
{
  __shared__ float As[GT_M][A_PAD];
  __shared__ float Bs[2][GT_K][B_PAD];

  const int b   = blockIdx.z;
  const int o0  = blockIdx.y * GT_M;
  const int n0  = blockIdx.x * GT_N;
  const int tid = threadIdx.x;
  const int lane = tid & 31;
  const int wave = tid >> 5;
  const int m_sub = (wave & 3) * 16;          // 0,16,32,48
  const int n_sub = (wave >> 2) * 32;         // 0,32

  const float* Xb = X + (size_t)b * Cin * NPIX + n0;

  v8f c0 = {};
  v8f c1 = {};

  const int l  = lane & 15;
  const int kh = (lane >> 4) << 1;            // 0 or 2 (K-half per A/B layout)
  const int nk = Cin / GT_K;

  // prologue: post TDM for B tile 0
  if (wave == 0) {
    tdm_load_tile_2d((uint32_t)(uintptr_t)&Bs[0][0][0], Xb,
                     NPIX, (uint32_t)Cin, GT_N, GT_K, NPIX);
  }

  for (int t = 0; t < nk; ++t) {
    const int k0 = t * GT_K;

    // ---- stage A tile: W[o0..o0+63][k0..k0+31]  (512 float4, 2/thread)
    #pragma unroll
    for (int i = tid; i < (GT_M * GT_K) / 4; i += 256) {
      int r = i >> 3;                // (i*4)/32
      int c = (i & 7) << 2;          // (i*4)%32
      float4 v = *(const float4*)(W + (size_t)(o0 + r) * Cin + k0 + c);
      As[r][c + 0] = v.x; As[r][c + 1] = v.y;
      As[r][c + 2] = v.z; As[r][c + 3] = v.w;
    }

    // ---- post TDM for B tile t+1 into the other buffer, then wait for tile t
    if (wave == 0) {
      if (t + 1 < nk) {
        tdm_load_tile_2d((uint32_t)(uintptr_t)&Bs[(t + 1) & 1][0][0],
                         Xb + (size_t)(k0 + GT_K) * NPIX,
                         NPIX, (uint32_t)(Cin - k0 - GT_K), GT_N, GT_K, NPIX);
        __builtin_amdgcn_s_wait_tensorcnt((short)1);  // in-order: tile t resident
      } else {
        __builtin_amdgcn_s_wait_tensorcnt((short)0);
      }
    }
    __syncthreads();

    const float (*Bt)[B_PAD] = Bs[t & 1];
    #pragma unroll
    for (int kk = 0; kk < GT_K; kk += 4) {
      v2f a, b0, b1;
      a.x  = As[m_sub + l][kk + kh];
      a.y  = As[m_sub + l][kk + kh + 1];
      b0.x = Bt[kk + kh][n_sub + l];
      b0.y = Bt[kk + kh + 1][n_sub + l];
      b1.x = Bt[kk + kh][n_sub + 16 + l];
      b1.y = Bt[kk + kh + 1][n_sub + 16 + l];
      c0 = __builtin_amdgcn_wmma_f32_16x16x4_f32(false, a, false, b0, (short)0, c0, false, false);
      c1 = __builtin_amdgcn_wmma_f32_16x16x4_f32(false, a, false, b1, (short)0, c1, false, false);
    }
    __syncthreads();
  }

  // ---- store: C/D layout: VGPR r, lanes 0-15 -> M=r, lanes 16-31 -> M=r+8
  const int rofs = (lane >> 4) * 8;
  float* Yb = Y + ((size_t)b * O + o0 + m_sub + rofs) * NPIX + n0 + n_sub;
  #pragma unroll
  for (int r = 0; r < 8; ++r) {
    Yb[(size_t)r * NPIX + l]      = c0[r];
    Yb[(size_t)r * NPIX + 16 + l] = c1[r];
  }
}

// ---------------------------------------------------------------------------
// Kernel 2: depthwise 3x3, SAME padding. One 16x16 tile per 256-thread block,
// 18x18 halo staged in LDS. blockIdx.z = b*C + c.
// ---------------------------------------------------------------------------
__global__ __launch_bounds__(256) void dwconv3x3_kernel(
    const float* __restrict__ X,   // [B*C, 128, 128]
    const float* __restrict__ Wd,  // [C, 9]
    float* __restrict__ Y,         // [B*C, 128, 128]
    int C)
{
  __shared__ float tile[18][19];
  const int bc = blockIdx.z;
  const int c  = bc % C;
  const float* Xc = X + (size_t)bc * NPIX;
  const int ty0 = blockIdx.y * 16;
  const int tx0 = blockIdx.x * 16;
  const int tx = threadIdx.x & 15;
  const int ty = threadIdx.x >> 4;

  for (int i = threadIdx.x; i < 18 * 18; i += 256) {
    int yy = i / 18 - 1 + ty0;
    int xx = i % 18 - 1 + tx0;
    float v = 0.0f;
    if (yy >= 0 && yy < IMGH && xx >= 0 && xx < IMGW) v = Xc[yy * IMGW + xx];
    tile[i / 18][i % 18] = v;
  }
  __syncthreads();

  const float* wk = Wd + (size_t)c * 9;
  float w0 = wk[0], w1 = wk[1], w2 = wk[2];
  float w3 = wk[3], w4 = wk[4], w5 = wk[5];
  float w6 = wk[6], w7 = wk[7], w8 = wk[8];

  float acc = tile[ty + 0][tx + 0] * w0 + tile[ty + 0][tx + 1] * w1 + tile[ty + 0][tx + 2] * w2
            + tile[ty + 1][tx + 0] * w3 + tile[ty + 1][tx + 1] * w4 + tile[ty + 1][tx + 2] * w5
            + tile[ty + 2][tx + 0] * w6 + tile[ty + 2][tx + 1] * w7 + tile[ty + 2][tx + 2] * w8;

  Y[(size_t)bc * NPIX + (ty0 + ty) * IMGW + tx0 + tx] = acc;
}

// ---------------------------------------------------------------------------
// Kernel 3: per-row reciprocal L2 norm: inv[b*384+c] = 1/max(||X[b,c,:]||, eps)
// cb = channels per batch in X's layout (384 for q buffer, 768 for kv buffer;
// only the first 384 channels of each batch are k rows).
// ---------------------------------------------------------------------------
__global__ __launch_bounds__(256) void rownorm_kernel(
    const float* __restrict__ X, float* __restrict__ inv, int cb)
{
  const int b = blockIdx.x / CDIM;
  const int c = blockIdx.x % CDIM;
  const float* p = X + ((size_t)b * cb + c) * NPIX;

  float s = 0.0f;
  for (int i = threadIdx.x * 4; i < NPIX; i += 256 * 4) {
    float4 v = *(const float4*)(p + i);
    s += v.x * v.x + v.y * v.y + v.z * v.z + v.w * v.w;
  }
  #pragma unroll
  for (int off = 16; off > 0; off >>= 1) s += __shfl_xor(s, off, 32);

  __shared__ float red[8];
  if ((threadIdx.x & 31) == 0) red[threadIdx.x >> 5] = s;
  __syncthreads();
  if (threadIdx.x == 0) {
    float t = 0.0f;
    #pragma unroll
    for (int i = 0; i < 8; ++i) t += red[i];
    inv[blockIdx.x] = 1.0f / fmaxf(sqrtf(t), 1e-12f);
  }
}

// ---------------------------------------------------------------------------
// Kernel 4: attn[b,h,i,j] = invq_i * invk_j * temp[h] * sum_n q_i[n]*k_j[n]
// One (b,h) per block. 288 threads = 9 waves = 3x3 grid of 16x16 WMMA tiles
// covering the 48x48 output. K = 16384, streamed in 64-wide LDS tiles.
// ---------------------------------------------------------------------------
#define AT_K 64
__global__ __launch_bounds__(288) void attn_gram_kernel(
    const float* __restrict__ Q,     // [B, 384, NPIX]
    const float* __restrict__ K,     // [B, 768, NPIX] (k = first 384 channels)
    const float* __restrict__ invq,  // [B*384]
    const float* __restrict__ invk,  // [B*384]
    const float* __restrict__ temp,  // [HEADS]
    float* __restrict__ attn)        // [B*HEADS, 48, 48]
{
  __shared__ float Qs[CH][AT_K + 4];
  __shared__ float Ks[CH][AT_K + 4];

  const int bh = blockIdx.x;
  const int b = bh / HEADS;
  const int h = bh % HEADS;
  const float* Qp = Q + ((size_t)b * CDIM + h * CH) * NPIX;
  const float* Kp = K + ((size_t)b * (2 * CDIM) + h * CH) * NPIX;

  const int tid = threadIdx.x;
  const int lane = tid & 31;
  const int wave = tid >> 5;
  const int m0 = (wave / 3) * 16;
  const int n0 = (wave % 3) * 16;
  const int l  = lane & 15;
  const int kh = (lane >> 4) << 1;

  v8f acc = {};

  for (int k0 = 0; k0 < NPIX; k0 += AT_K) {
    for (int i = tid; i < (CH * AT_K) / 4; i += 288) {
      int r = i >> 4;                 // (i*4)/64
      int c = (i & 15) << 2;          // (i*4)%64
      float4 qv = *(const float4*)(Qp + (size_t)r * NPIX + k0 + c);
      float4 kv = *(const float4*)(Kp + (size_t)r * NPIX + k0 + c);
      Qs[r][c + 0] = qv.x; Qs[r][c + 1] = qv.y; Qs[r][c + 2] = qv.z; Qs[r][c + 3] = qv.w;
      Ks[r][c + 0] = kv.x; Ks[r][c + 1] = kv.y; Ks[r][c + 2] = kv.z; Ks[r][c + 3] = kv.w;
    }
    __syncthreads();

    #pragma unroll
    for (int kk = 0; kk < AT_K; kk += 4) {
      v2f a, bv;
      a.x  = Qs[m0 + l][kk + kh];
      a.y  = Qs[m0 + l][kk + kh + 1];
      bv.x = Ks[n0 + l][kk + kh];       // B[K][N] = k_row(N) at pixel(K): transpose via LDS
      bv.y = Ks[n0 + l][kk + kh + 1];
      acc = __builtin_amdgcn_wmma_f32_16x16x4_f32(false, a, false, bv, (short)0, acc, false, false);
    }
    __syncthreads();
  }

  const int rofs = (lane >> 4) * 8;
  const float tp = temp[h];
  float* A = attn + (size_t)bh * CH * CH;
  #pragma unroll
  for (int r = 0; r < 8; ++r) {
    int row = m0 + rofs + r;
    int col = n0 + l;
    A[row * CH + col] = acc[r] * invq[b * CDIM + h * CH + row]
                               * invk[b * CDIM + h * CH + col] * tp;
  }
}

// ---------------------------------------------------------------------------
// Kernel 5: combined top-m softmax.
// acomb[row,:] = sum_i a_mix[i] * softmax(mask_topm_i(attn[row,:]))
// m in {24, 32, 36, 38} (= C/2, 2C/3, 3C/4, 4C/5 for C=48).
// One thread per row (48 rows per (b,h)); tiny kernel, selection by full sort.
// ---------------------------------------------------------------------------
__global__ __launch_bounds__(64) void topm_softmax_kernel(
    const float* __restrict__ attn, const float* __restrict__ amix,
    float* __restrict__ acomb)
{
  const int bh  = blockIdx.x;
  const int row = threadIdx.x;
  if (row >= CH) return;

  const float* a = attn + (size_t)bh * CH * CH + row * CH;
  float v[CH], s[CH], e[CH], o[CH];
  #pragma unroll
  for (int j = 0; j < CH; ++j) { v[j] = a[j]; s[j] = v[j]; o[j] = 0.0f; }

  // insertion sort, descending
  for (int i = 1; i < CH; ++i) {
    float key = s[i];
    int j = i - 1;
    while (j >= 0 && s[j] < key) { s[j + 1] = s[j]; --j; }
    s[j + 1] = key;
  }

  const float mx = s[0];
  #pragma unroll
  for (int j = 0; j < CH; ++j) e[j] = expf(v[j] - mx);

  const int ms[4] = { CH / 2, (2 * CH) / 3, (3 * CH) / 4, (4 * CH) / 5 };
  for (int mi = 0; mi < 4; ++mi) {
    float thr = s[ms[mi] - 1];
    float denom = 0.0f;
    for (int j = 0; j < CH; ++j) if (v[j] >= thr) denom += e[j];
    float scale = amix[mi] / denom;
    for (int j = 0; j < CH; ++j) if (v[j] >= thr) o[j] += scale * e[j];
  }

  float* out = acomb + (size_t)bh * CH * CH + row * CH;
  #pragma unroll
  for (int j = 0; j < CH; ++j) out[j] = o[j];
}

// ---------------------------------------------------------------------------
// Kernel 6: out[b, h*48+i, n] = sum_j acomb[b,h,i,j] * v[b,h,j,n]
// 384 threads = 12 waves arranged 3(M) x 4(N): 48 x 64 output tile, K = 48.
// ---------------------------------------------------------------------------
__global__ __launch_bounds__(384) void av_wmma_kernel(
    const float* __restrict__ Ac,   // [B*HEADS, 48, 48]
    const float* __restrict__ V,    // [B, 768, NPIX], v = channels 384..767
    float* __restrict__ Y)          // [B, 384, NPIX]
{
  __shared__ float As[CH][52];
  __shared__ float Bs[CH][68];

  const int bh = blockIdx.y;
  const int b = bh / HEADS;
  const int h = bh % HEADS;
  const int n0 = blockIdx.x * 64;
  const int tid = threadIdx.x;
  const int lane = tid & 31;
  const int wave = tid >> 5;
  const int m0 = (wave >> 2) * 16;   // 0,16,32
  const int nn = (wave & 3) * 16;    // 0,16,32,48

  const float* Ap = Ac + (size_t)bh * CH * CH;
  for (int i = tid; i < CH * CH; i += 384) As[i / CH][i % CH] = Ap[i];

  const float* Vp = V + ((size_t)b * (2 * CDIM) + CDIM + h * CH) * NPIX + n0;
  for (int i = tid; i < (CH * 64) / 4; i += 384) {
    int r = i >> 4;
    int c = (i & 15) << 2;
    float4 v = *(const float4*)(Vp + (size_t)r * NPIX + c);
    Bs[r][c + 0] = v.x; Bs[r][c + 1] = v.y; Bs[r][c + 2] = v.z; Bs[r][c + 3] = v.w;
  }
  __syncthreads();

  const int l  = lane & 15;
  const int kh = (lane >> 4) << 1;
  v8f acc = {};
  #pragma unroll
  for (int kk = 0; kk < CH; kk += 4) {
    v2f a, bv;
    a.x  = As[m0 + l][kk + kh];
    a.y  = As[m0 + l][kk + kh + 1];
    bv.x = Bs[kk + kh][nn + l];
    bv.y = Bs[kk + kh + 1][nn + l];
    acc = __builtin_amdgcn_wmma_f32_16x16x4_f32(false, a, false, bv, (short)0, acc, false, false);
  }

  const int rofs = (lane >> 4) * 8;
  float* Yp = Y + ((size_t)b * CDIM + h * CH + m0 + rofs) * NPIX + n0 + nn;
  #pragma unroll
  for (int r = 0; r < 8; ++r) Yp[(size_t)r * NPIX + l] = acc[r];
}

// ---------------------------------------------------------------------------
// Launch
// ---------------------------------------------------------------------------
extern "C" void kernel_launch(void* const* d_in, const int* in_sizes, int n_in,
                              void* d_out, int out_size, void* d_ws, size_t ws_size,
                              hipStream_t stream) {
  const float* x_q    = (const float*)d_in[0];   // [8,384,128,128]
  const float* x_kv   = (const float*)d_in[1];   // [8,384,128,128]
  const float* q_w    = (const float*)d_in[2];   // [384,384]
  const float* q_dw   = (const float*)d_in[3];   // [384,1,3,3]
  const float* kv_w   = (const float*)d_in[4];   // [768,384]
  const float* kv_dw  = (const float*)d_in[5];   // [768,1,3,3]
  const float* proj_w = (const float*)d_in[6];   // [384,384]
  const float* temp   = (const float*)d_in[7];   // [8,1,1]
  const float* a_mix  = (const float*)d_in[8];   // [4]
  float* out = (float*)d_out;

  float* ws = (float*)d_ws;
  const size_t SZ = (size_t)BATCH * CDIM * NPIX;       // 50,331,648 floats
  float* f_pre  = ws;                                   // 2*SZ (conv1x1 staging)
  float* f_q    = ws + 2 * SZ;                          // SZ   (q after dwconv)
  float* f_kv   = ws + 3 * SZ;                          // 2*SZ (k then v per batch)
  float* f_out  = ws + 5 * SZ;                          // SZ   (attn output)
  float* f_invq = ws + 6 * SZ;                          // 3072
  float* f_invk = f_invq + BATCH * CDIM;                // 3072
  float* f_attn  = f_invk + BATCH * CDIM;               // 64*48*48
  float* f_acomb = f_attn + BATCH * HEADS * CH * CH;    // 64*48*48

  // 1) q = dwconv3x3(conv1x1(x_q, q_w), q_dw)
  gemm1x1_wmma_kernel<<<dim3(NPIX / GT_N, CDIM / GT_M, BATCH), 256, 0, stream>>>(
      x_q, q_w, f_pre, CDIM, CDIM);
  dwconv3x3_kernel<<<dim3(IMGW / 16, IMGH / 16, BATCH * CDIM), 256, 0, stream>>>(
      f_pre, q_dw, f_q, CDIM);

  // 2) kv = dwconv3x3(conv1x1(x_kv, kv_w), kv_dw)
  gemm1x1_wmma_kernel<<<dim3(NPIX / GT_N, (2 * CDIM) / GT_M, BATCH), 256, 0, stream>>>(
      x_kv, kv_w, f_pre, CDIM, 2 * CDIM);
  dwconv3x3_kernel<<<dim3(IMGW / 16, IMGH / 16, BATCH * 2 * CDIM), 256, 0, stream>>>(
      f_pre, kv_dw, f_kv, 2 * CDIM);

  // 3) reciprocal L2 norms of q and k rows (norm folded into attn epilogue)
  rownorm_kernel<<<BATCH * CDIM, 256, 0, stream>>>(f_q, f_invq, CDIM);
  rownorm_kernel<<<BATCH * CDIM, 256, 0, stream>>>(f_kv, f_invk, 2 * CDIM);

  // 4) attn = (q_n k_n^T) * temperature  [per (b,h): 48x48, K=16384]
  attn_gram_kernel<<<BATCH * HEADS, 288, 0, stream>>>(
      f_q, f_kv, f_invq, f_invk, temp, f_attn);

  // 5) combined top-m softmax: acomb = sum_i a_mix[i]*softmax(mask_i(attn))
  topm_softmax_kernel<<<BATCH * HEADS, 64, 0, stream>>>(f_attn, a_mix, f_acomb);

  // 6) out = acomb @ v
  av_wmma_kernel<<<dim3(NPIX / 64, BATCH * HEADS), 384, 0, stream>>>(
      f_acomb, f_kv, f_out);

  // 7) final = conv1x1(out, proj_w)
  gemm1x1_wmma_kernel<<<dim3(NPIX / GT_N, CDIM / GT_M, BATCH), 256, 0, stream>>>(
      f_out, proj_w, out, CDIM, CDIM);
}